// Head_24455543783706
// MI455X (gfx1250) — compile-verified
//
#include <hip/hip_runtime.h>
#include <math.h>

typedef float v2f __attribute__((ext_vector_type(2)));
typedef float v4f __attribute__((ext_vector_type(4)));
typedef float v8f __attribute__((ext_vector_type(8)));

#define BB   64
#define NN   16384
#define MM   128
#define CC   1024
#define EPSF 1e-16f
#define WLEN (BB * NN)

// workspace layout (floats)
#define WS_K     0        // BB*MM = 8192
#define WS_KNORM 8192     // BB
#define WS_PAR   8256     // BB*6 : beta, gate, gamma, s0, s1, s2
#define WS_MAX   8640     // BB
#define WS_DEN   8704     // BB
#define WS_WPSUM 8768     // BB
#define WS_S     16384    // BB*NN = 1048576

__device__ __forceinline__ float softplusf(float x) {
    return (x > 20.f) ? x : log1pf(expf(x));
}

// ---------------------------------------------------------------------------
// Kernel 1: k = tanh(ctrl @ key_w + key_b) via V_WMMA_F32_16X16X4_F32.
// One wave per 16x16 output tile; 256 K-steps of 4 (full f32 precision).
// A layout (16x4 f32): lanes 0-15 -> M=lane, K={k0,k0+1}; lanes 16-31 -> K={k0+2,k0+3}
// B layout: transpose-symmetric (lane = N column, half-wave selects K pair)
// D layout (16x16 f32): VGPR r, lanes 0-15 -> M=r, N=lane; lanes 16-31 -> M=r+8
// ---------------------------------------------------------------------------
__global__ __launch_bounds__(32)
void k_proj_key(const float* __restrict__ ctrl, const float* __restrict__ key_w,
                const float* __restrict__ key_b, float* __restrict__ ws) {
    const int lane = threadIdx.x;
    const int m0 = blockIdx.x * 16;   // 8 tiles over M=128
    const int b0 = blockIdx.y * 16;   // 4 tiles over B=64
    const int row = lane & 15;
    const int kb  = (lane >> 4) * 2;

    v8f acc = {};
    const float* actrl = ctrl + (size_t)(b0 + row) * CC;
    for (int k0 = 0; k0 < CC; k0 += 4) {
        v2f a, b;
        a.x = actrl[k0 + kb];
        a.y = actrl[k0 + kb + 1];
        b.x = key_w[(size_t)(k0 + kb)     * MM + m0 + row];
        b.y = key_w[(size_t)(k0 + kb + 1) * MM + m0 + row];
        acc = __builtin_amdgcn_wmma_f32_16x16x4_f32(
            /*neg_a=*/false, a, /*neg_b=*/false, b,
            /*c_mod=*/(short)0, acc, /*reuse_a=*/false, /*reuse_b=*/false);
    }
    const int half = lane >> 4;
    const int ncol = lane & 15;
    const float bias = key_b[m0 + ncol];
    float* kout = ws + WS_K;
#pragma unroll
    for (int r = 0; r < 8; ++r) {
        int brow = b0 + r + 8 * half;
        kout[brow * MM + m0 + ncol] = tanhf(acc[r] + bias);
    }
}

// ---------------------------------------------------------------------------
// Kernel 2: per-batch scalar projections + ||k|| + init accumulators.
// ---------------------------------------------------------------------------
__global__ __launch_bounds__(128)
void k_scalars(const float* __restrict__ ctrl,
               const float* __restrict__ beta_w,  const float* __restrict__ beta_b,
               const float* __restrict__ gate_w,  const float* __restrict__ gate_b,
               const float* __restrict__ shift_w, const float* __restrict__ shift_b,
               const float* __restrict__ gamma_w, const float* __restrict__ gamma_b,
               float* __restrict__ ws, float* __restrict__ out) {
    const int b = blockIdx.x;
    const int t = threadIdx.x;
    float pb = 0.f, pg = 0.f, pm = 0.f, p0 = 0.f, p1 = 0.f, p2 = 0.f;
    const float* cr = ctrl + (size_t)b * CC;
    for (int c = t; c < CC; c += 128) {
        float x = cr[c];
        pb += x * beta_w[c];
        pg += x * gate_w[c];
        pm += x * gamma_w[c];
        p0 += x * shift_w[c * 3 + 0];
        p1 += x * shift_w[c * 3 + 1];
        p2 += x * shift_w[c * 3 + 2];
    }
    float kk = ws[WS_K + b * MM + t];
    float pk = kk * kk;

    __shared__ float red[7][128];
    red[0][t] = pb; red[1][t] = pg; red[2][t] = pm;
    red[3][t] = p0; red[4][t] = p1; red[5][t] = p2; red[6][t] = pk;
    __syncthreads();
    for (int s = 64; s > 0; s >>= 1) {
        if (t < s)
            for (int j = 0; j < 7; ++j) red[j][t] += red[j][t + s];
        __syncthreads();
    }
    // init read_vec output region + wp-sum accumulator (harness doesn't re-zero)
    out[(size_t)WLEN + (size_t)b * MM + t] = 0.f;
    if (t == 0) {
        ws[WS_WPSUM + b] = 0.f;
        ws[WS_PAR + b * 6 + 0] = softplusf(red[0][0] + beta_b[0]);            // beta
        ws[WS_PAR + b * 6 + 1] = 1.f / (1.f + expf(-(red[1][0] + gate_b[0]))); // gate
        ws[WS_PAR + b * 6 + 2] = 1.f + softplusf(red[2][0] + gamma_b[0]);     // gamma
        float x0 = red[3][0] + shift_b[0];
        float x1 = red[4][0] + shift_b[1];
        float x2 = red[5][0] + shift_b[2];
        float mx = fmaxf(x0, fmaxf(x1, x2));
        float e0 = expf(x0 - mx), e1 = expf(x1 - mx), e2 = expf(x2 - mx);
        float inv = 1.f / (e0 + e1 + e2);
        ws[WS_PAR + b * 6 + 3] = e0 * inv;
        ws[WS_PAR + b * 6 + 4] = e1 * inv;
        ws[WS_PAR + b * 6 + 5] = e2 * inv;
        ws[WS_KNORM + b] = sqrtf(red[6][0]);
    }
}

// ---------------------------------------------------------------------------
// Kernel 3: streaming pass 1 — s[b,n] = beta * cosine(k_b, memory[b,n]).
// Lane-per-slot: each thread streams its own 512B row (32x B128), k staged in
// LDS (broadcast reads). No cross-lane reductions -> ~2x fewer instructions
// per byte than a wave-cooperative scheme; stays bandwidth-bound.
// ---------------------------------------------------------------------------
__global__ __launch_bounds__(256)
void k_dots(const float* __restrict__ mem, float* __restrict__ ws) {
    const int b = blockIdx.y;
    const int n = blockIdx.x * 256 + threadIdx.x;   // slot owned by this thread

    __shared__ float sk[MM];
    if (threadIdx.x < 32)
        *(v4f*)&sk[threadIdx.x * 4] =
            *(const v4f*)(ws + WS_K + b * MM + threadIdx.x * 4);
    __syncthreads();

    const float beta  = ws[WS_PAR + b * 6 + 0];
    const float knorm = ws[WS_KNORM + b];
    const float* row = mem + ((size_t)b * NN + n) * MM;

    float d = 0.f, sq = 0.f;
#pragma unroll 8
    for (int j = 0; j < 32; ++j) {
        v4f mv = __builtin_nontemporal_load((const v4f*)(row + j * 4));
        v4f kc = *(const v4f*)&sk[j * 4];
        d  += mv.x * kc.x + mv.y * kc.y + mv.z * kc.z + mv.w * kc.w;
        sq += mv.x * mv.x + mv.y * mv.y + mv.z * mv.z + mv.w * mv.w;
    }
    float sim = d / (knorm * sqrtf(sq) + EPSF);
    ws[WS_S + (size_t)b * NN + n] = beta * sim;    // coalesced B32 store
}

// ---------------------------------------------------------------------------
// Kernel 4: per-batch softmax stats (max, sum-exp) over N. s[] is L2-hot.
// ---------------------------------------------------------------------------
__global__ __launch_bounds__(256)
void k_softmax_stats(float* __restrict__ ws) {
    const int b = blockIdx.x, t = threadIdx.x;
    const float* s = ws + WS_S + (size_t)b * NN;
    __shared__ float red[256];
    float mx = -3.0e38f;
    for (int n = t; n < NN; n += 256) mx = fmaxf(mx, s[n]);
    red[t] = mx; __syncthreads();
    for (int st = 128; st > 0; st >>= 1) {
        if (t < st) red[t] = fmaxf(red[t], red[t + st]);
        __syncthreads();
    }
    mx = red[0]; __syncthreads();
    float sm = 0.f;
    for (int n = t; n < NN; n += 256) sm += expf(s[n] - mx);
    red[t] = sm; __syncthreads();
    for (int st = 128; st > 0; st >>= 1) {
        if (t < st) red[t] += red[t + st];
        __syncthreads();
    }
    if (t == 0) { ws[WS_MAX + b] = mx; ws[WS_DEN + b] = red[0]; }
}

// ---------------------------------------------------------------------------
// Kernel 5: gated interpolation + circular 3-tap shift + w^gamma.
// Stores wp into the weights output region; accumulates sum(wp) per batch.
// ---------------------------------------------------------------------------
__global__ __launch_bounds__(256)
void k_shift_pow(const float* __restrict__ prev, float* __restrict__ ws,
                 float* __restrict__ out) {
    const int b = blockIdx.y;
    const int n = blockIdx.x * 256 + threadIdx.x;
    const float gate  = ws[WS_PAR + b * 6 + 1];
    const float gamma = ws[WS_PAR + b * 6 + 2];
    const float s0 = ws[WS_PAR + b * 6 + 3];
    const float s1 = ws[WS_PAR + b * 6 + 4];
    const float s2 = ws[WS_PAR + b * 6 + 5];
    const float mx = ws[WS_MAX + b];
    const float invden = 1.f / ws[WS_DEN + b];
    const float* sarr = ws + WS_S + (size_t)b * NN;
    const float* parr = prev + (size_t)b * NN;

    const int nm1 = (n + NN - 1) & (NN - 1);
    const int np1 = (n + 1) & (NN - 1);
    auto wfun = [&](int j) {
        return gate * expf(sarr[j] - mx) * invden + (1.f - gate) * parr[j];
    };
    // roll(w,+1)[n] = w[n-1]; roll(w,-1)[n] = w[n+1]
    float wsh = s0 * wfun(nm1) + s1 * wfun(n) + s2 * wfun(np1);
    float wp  = powf(wsh, gamma);
    out[(size_t)b * NN + n] = wp;

    __shared__ float red[256];
    red[threadIdx.x] = wp; __syncthreads();
    for (int st = 128; st > 0; st >>= 1) {
        if (threadIdx.x < st) red[threadIdx.x] += red[threadIdx.x + st];
        __syncthreads();
    }
    if (threadIdx.x == 0) atomicAdd(ws + WS_WPSUM + b, red[0]);
}

// ---------------------------------------------------------------------------
// Kernel 6: streaming pass 2 — normalize weights in place, accumulate
// read_vec = sum_n weights * memory[b,n,:]. Wave partials -> LDS -> atomics.
// Wave-per-row here because the m-accumulator must stay in 4 VGPRs per lane.
// ---------------------------------------------------------------------------
__global__ __launch_bounds__(256)
void k_read(const float* __restrict__ mem, float* __restrict__ out,
            const float* __restrict__ ws) {
    const int b    = blockIdx.y;
    const int lane = threadIdx.x & 31;
    const int wv   = threadIdx.x >> 5;
    const int nbase = blockIdx.x * 256 + wv * 32;
    const float inv = 1.f / (ws[WS_WPSUM + b] + EPSF);
    float* wout = out + (size_t)b * NN;
    const float* mrow = mem + ((size_t)b * NN + nbase) * MM;

    v4f acc = {0.f, 0.f, 0.f, 0.f};
    float mywt = 0.f;
    for (int i = 0; i < 32; ++i) {
        float wt = wout[nbase + i] * inv;   // wp was staged here by kernel 5
        if (i == lane) mywt = wt;
        v4f mv = __builtin_nontemporal_load(
            (const v4f*)(mrow + (size_t)i * MM + lane * 4));
        acc.x += wt * mv.x; acc.y += wt * mv.y;
        acc.z += wt * mv.z; acc.w += wt * mv.w;
    }
    wout[nbase + lane] = mywt;              // one coalesced normalized-weight store

    __shared__ float racc[8 * 128];
    *(v4f*)&racc[wv * 128 + lane * 4] = acc;
    __syncthreads();
    const int t = threadIdx.x;
    if (t < 128) {
        float s = 0.f;
        for (int w2 = 0; w2 < 8; ++w2) s += racc[w2 * 128 + t];
        atomicAdd(out + (size_t)WLEN + (size_t)b * MM + t, s);
    }
}

// ---------------------------------------------------------------------------
extern "C" void kernel_launch(void* const* d_in, const int* in_sizes, int n_in,
                              void* d_out, int out_size, void* d_ws, size_t ws_size,
                              hipStream_t stream) {
    const float* ctrl    = (const float*)d_in[0];
    const float* prev    = (const float*)d_in[1];
    const float* memory  = (const float*)d_in[2];
    const float* key_w   = (const float*)d_in[3];
    const float* key_b   = (const float*)d_in[4];
    const float* beta_w  = (const float*)d_in[5];
    const float* beta_b  = (const float*)d_in[6];
    const float* gate_w  = (const float*)d_in[7];
    const float* gate_b  = (const float*)d_in[8];
    const float* shift_w = (const float*)d_in[9];
    const float* shift_b = (const float*)d_in[10];
    const float* gamma_w = (const float*)d_in[11];
    const float* gamma_b = (const float*)d_in[12];
    float* out = (float*)d_out;
    float* ws  = (float*)d_ws;

    k_proj_key<<<dim3(MM / 16, BB / 16), 32, 0, stream>>>(ctrl, key_w, key_b, ws);
    k_scalars<<<BB, 128, 0, stream>>>(ctrl, beta_w, beta_b, gate_w, gate_b,
                                      shift_w, shift_b, gamma_w, gamma_b, ws, out);
    k_dots<<<dim3(NN / 256, BB), 256, 0, stream>>>(memory, ws);
    k_softmax_stats<<<BB, 256, 0, stream>>>(ws);
    k_shift_pow<<<dim3(NN / 256, BB), 256, 0, stream>>>(prev, ws, out);
    k_read<<<dim3(NN / 256, BB), 256, 0, stream>>>(memory, out, ws);
}